// lr_RNN_44762149159371
// MI455X (gfx1250) — compile-verified
//
#include <hip/hip_runtime.h>
#include <hip/hip_bf16.h>
#include <cmath>

#define B_   32
#define T_   2048
#define P_   128
#define K_   256
#define R_   8
#define M_   64
#define NTOK (B_ * T_)   // 65536 tokens

typedef __attribute__((ext_vector_type(2))) float v2f;
typedef __attribute__((ext_vector_type(8))) float v8f;

#define LDP 68   // LDS row pitch (floats): 16B-aligned for b128 staging,
                 // conflict-free for v2f fragment reads (banks 4r+k+{0..3})

// ---- CDNA5 async memory->LDS (ASYNCcnt-tracked, no data VGPRs) -----------
__device__ __forceinline__ void async_ld_b128(unsigned lds_addr, const float* gaddr) {
  asm volatile("global_load_async_to_lds_b128 %0, %1, off"
               :: "v"(lds_addr), "v"(gaddr) : "memory");
}
__device__ __forceinline__ void wait_async0() {
  asm volatile("s_wait_asynccnt 0x0" ::: "memory");
}

// ---------------------------------------------------------------------------
// C[n][o] = act( sum_k A[n][k] * W[o][k] + bias[o] )    (A @ W^T, row-major)
// Block = 128 threads = 4 waves; each wave owns one 16x16 tile via
// V_WMMA_F32_16X16X4_F32. K streamed in 64-wide chunks through ping-pong LDS
// buffers filled by GLOBAL_LOAD_ASYNC_TO_LDS_B128: the next chunk's DMA
// overlaps the current chunk's 16 WMMAs; s_wait_asynccnt 0 + barrier flips.
// ---------------------------------------------------------------------------
template<int KIN>
__global__ __launch_bounds__(128) void gemm_bias_act(
    const float* __restrict__ A, const float* __restrict__ W,
    const float* __restrict__ bias, float* __restrict__ C,
    int Kout, int act)
{
  __shared__ __align__(16) float As[2][16 * LDP];
  __shared__ __align__(16) float Ws[2][64 * LDP];

  const int tid  = threadIdx.x;
  const int wave = tid >> 5;
  const int lane = tid & 31;
  const int row0 = blockIdx.x * 16;
  const int col0 = blockIdx.y * 64;

  const int rr   = lane & 15;          // A-row / B-col within the 16x16 tile
  const int koff = (lane >> 4) * 2;    // lanes 0-15: K={k,k+1}; 16-31: K={k+2,k+3}

  // Staging: float4 per slot; slot rows step by 8 -> one base + immediates.
  // A tile 16x64 = 2 slots/thread, W tile 64x64 = 8 slots/thread.
  const int sr = tid >> 4;             // staging row base (0..7)
  const int sc = (tid & 15) * 4;       // staging col (float index)

  const float* aBase = A + (size_t)(row0 + sr) * KIN + sc;
  const float* wBase = W + (size_t)(col0 + sr) * KIN + sc;
  unsigned ldsA[2], ldsW[2];
#pragma unroll
  for (int i = 0; i < 2; ++i) {
    ldsA[i] = (unsigned)(size_t)&As[i][sr * LDP + sc];
    ldsW[i] = (unsigned)(size_t)&Ws[i][sr * LDP + sc];
  }

  // ---- issue chunk 0 into buffer 0 ----
  async_ld_b128(ldsA[0],                 aBase);
  async_ld_b128(ldsA[0] + 8 * LDP * 4,   aBase + (size_t)8 * KIN);
#pragma unroll
  for (int j = 0; j < 8; ++j)
    async_ld_b128(ldsW[0] + j * 8 * LDP * 4, wBase + (size_t)j * 8 * KIN);
  wait_async0();
  __syncthreads();

  v8f acc = {0.f, 0.f, 0.f, 0.f, 0.f, 0.f, 0.f, 0.f};

  int ibuf = 0;
  for (int kc = 0; kc < KIN; kc += 64) {
    // issue next chunk's async DMA into the other buffer
    if (kc + 64 < KIN) {
      const int nb = ibuf ^ 1;
      const float* an = aBase + kc + 64;
      const float* wn = wBase + kc + 64;
      async_ld_b128(ldsA[nb],               an);
      async_ld_b128(ldsA[nb] + 8 * LDP * 4, an + (size_t)8 * KIN);
#pragma unroll
      for (int j = 0; j < 8; ++j)
        async_ld_b128(ldsW[nb] + j * 8 * LDP * 4, wn + (size_t)j * 8 * KIN);
    }

    const float* as = &As[ibuf][rr * LDP + koff];
    const float* ws = &Ws[ibuf][(wave * 16 + rr) * LDP + koff];
#pragma unroll
    for (int k = 0; k < 64; k += 4) {
      v2f a = *(const v2f*)&as[k];
      v2f b = *(const v2f*)&ws[k];
      acc = __builtin_amdgcn_wmma_f32_16x16x4_f32(
          false, a, false, b, (short)0, acc, false, false);
    }

    wait_async0();       // own async writes landed in LDS
    __syncthreads();     // everyone's writes landed, everyone done reading
    ibuf ^= 1;
  }

  // C/D layout: VGPR i -> lanes 0-15 = (M=i, N=lane); lanes 16-31 = (M=i+8, N=lane-16)
  const int col   = col0 + wave * 16 + rr;
  const float bv  = bias[col];
  const int mbase = (lane < 16) ? 0 : 8;
#pragma unroll
  for (int i = 0; i < 8; ++i) {
    float v = acc[i] + bv;
    if (act) v = tanhf(v);
    C[(size_t)(row0 + mbase + i) * Kout + col] = v;
  }
}

// ---------------------------------------------------------------------------
// Serial recurrence. One wave32 per batch row (32 single-wave workgroups):
// barriers degenerate to S_NOP, LDS ops are in-order within the wave.
// State: x (256 f32 -> 8/lane), z (64 f32 -> 2/lane). Weights live in LDS.
// Emits x-state history (decoded afterwards by batched WMMA GEMMs) + z_pred.
// ---------------------------------------------------------------------------
__global__ __launch_bounds__(32) void scan_kernel(
    const float* __restrict__ xseq,   // (B,T,K) ws
    const float* __restrict__ zseq,   // (B,T,M) = d_out region 3
    const float* __restrict__ Wz,     // (M,M)
    const float* __restrict__ Azw,    // (R,M)
    const float* __restrict__ Azb,    // (R)
    const float* __restrict__ Lm,     // (K,R)
    const float* __restrict__ Rm,     // (K,R)
    float* __restrict__ xhist,        // (B,T,K) ws
    float* __restrict__ zpred)        // (B,T,M) = d_out region 2
{
  const int b    = blockIdx.x;
  const int lane = threadIdx.x;

  __shared__ float sWz[64 * 65];       // padded rows: conflict-free row reads
  __shared__ float sAz[8 * 64];
  __shared__ float sAzb[8];
  __shared__ float sL[256 * 8];
  __shared__ float sRm[256 * 8];
  __shared__ float tzbuf[64], zbuf[64], sbuf[8], txbuf[256], ubuf[32], vbuf[8];

  for (int i = lane; i < 64 * 64; i += 32) sWz[(i >> 6) * 65 + (i & 63)] = Wz[i];
  for (int i = lane; i < 8 * 64;  i += 32) sAz[i] = Azw[i];
  if (lane < 8) sAzb[lane] = Azb[lane];
  for (int i = lane; i < 256 * 8; i += 32) { sL[i] = Lm[i]; sRm[i] = Rm[i]; }

  const size_t xb = (size_t)b * T_ * K_;
  const size_t zb = (size_t)b * T_ * M_;

  float x[8], z0, z1;
#pragma unroll
  for (int i = 0; i < 8; ++i) x[i] = xseq[xb + lane + 32 * i];
  z0 = zseq[zb + lane];
  z1 = zseq[zb + lane + 32];
  __syncthreads();

  for (int t = 0; t < T_; ++t) {
    if (t > 0 && (t % 5) == 0) {   // TAU = 5 reset
      const size_t xo = xb + (size_t)t * K_;
      const size_t zo = zb + (size_t)t * M_;
#pragma unroll
      for (int i = 0; i < 8; ++i) x[i] = xseq[xo + lane + 32 * i];
      z0 = zseq[zo + lane];
      z1 = zseq[zo + lane + 32];
    }

    // ---- z update: z += (tanh(z) @ Wz^T - z) / TAU_Z ----
    tzbuf[lane]      = tanhf(z0);
    tzbuf[lane + 32] = tanhf(z1);
    __syncthreads();
    float a0 = 0.f, a1 = 0.f;
#pragma unroll 8
    for (int j = 0; j < 64; ++j) {
      float tzj = tzbuf[j];
      a0 += tzj * sWz[lane * 65 + j];
      a1 += tzj * sWz[(lane + 32) * 65 + j];
    }
    z0 += (a0 - z0) * 0.01f;
    z1 += (a1 - z1) * 0.01f;
    zbuf[lane] = z0; zbuf[lane + 32] = z1;
    __syncthreads();

    // ---- s = sigmoid(z @ Az_w^T + Az_b) ----
    if (lane < 8) {
      float acc = sAzb[lane];
      for (int j = 0; j < 64; ++j) acc += zbuf[j] * sAz[lane * 64 + j];
      sbuf[lane] = 1.f / (1.f + __expf(-acc));
    }
#pragma unroll
    for (int i = 0; i < 8; ++i) txbuf[lane + 32 * i] = tanhf(x[i]);
    __syncthreads();

    // ---- u = tanh(x) @ L : 32 lanes = 8 r-values x 4 k-segments ----
    {
      int r = lane & 7, seg = lane >> 3;
      float up = 0.f;
      for (int k = seg * 64; k < seg * 64 + 64; ++k) up += txbuf[k] * sL[k * 8 + r];
      ubuf[lane] = up;
    }
    __syncthreads();
    if (lane < 8) {
      float u = ubuf[lane] + ubuf[lane + 8] + ubuf[lane + 16] + ubuf[lane + 24];
      vbuf[lane] = u * sbuf[lane];
    }
    __syncthreads();

    // ---- x += ((u*s) @ Rm^T - x) / TAU_X ----
#pragma unroll
    for (int i = 0; i < 8; ++i) {
      int k = lane + 32 * i;
      float dx = 0.f;
#pragma unroll
      for (int r = 0; r < 8; ++r) dx += vbuf[r] * sRm[k * 8 + r];
      x[i] += (dx - x[i]) * 0.01f;
    }

    // ---- emit x history + z_pred ----
    const size_t xo = xb + (size_t)t * K_;
    const size_t zo = zb + (size_t)t * M_;
#pragma unroll
    for (int i = 0; i < 8; ++i) xhist[xo + lane + 32 * i] = x[i];
    zpred[zo + lane]      = z0;
    zpred[zo + lane + 32] = z1;
    __syncthreads();
  }
}

// ---------------------------------------------------------------------------
extern "C" void kernel_launch(void* const* d_in, const int* in_sizes, int n_in,
                              void* d_out, int out_size, void* d_ws, size_t ws_size,
                              hipStream_t stream)
{
  (void)in_sizes; (void)n_in; (void)out_size; (void)ws_size;

  const float* in_seq = (const float*)d_in[0];
  const float* Lm     = (const float*)d_in[1];
  const float* Rm     = (const float*)d_in[2];
  const float* Wz     = (const float*)d_in[3];
  const float* Azw    = (const float*)d_in[4];
  const float* Azb    = (const float*)d_in[5];
  const float* ex1w = (const float*)d_in[6];  const float* ex1b = (const float*)d_in[7];
  const float* ex2w = (const float*)d_in[8];  const float* ex2b = (const float*)d_in[9];
  const float* ex3w = (const float*)d_in[10]; const float* ex3b = (const float*)d_in[11];
  const float* ez1w = (const float*)d_in[12]; const float* ez1b = (const float*)d_in[13];
  const float* ez2w = (const float*)d_in[14]; const float* ez2b = (const float*)d_in[15];
  const float* ez3w = (const float*)d_in[16]; const float* ez3b = (const float*)d_in[17];
  const float* de1w = (const float*)d_in[18]; const float* de1b = (const float*)d_in[19];
  const float* de2w = (const float*)d_in[20]; const float* de2b = (const float*)d_in[21];

  // Outputs concatenated: x_pred (B,T,P) | x_recon (B,T,P) | z_pred (B,T,M) | z_seq (B,T,M)
  float* out      = (float*)d_out;
  float* x_pred   = out;
  float* x_recon  = out + (size_t)NTOK * P_;
  float* z_pred   = out + (size_t)2 * NTOK * P_;
  float* z_seq    = z_pred + (size_t)NTOK * M_;

  // Workspace layout (floats): bufA | bufB | xseq | zA | zB   (~224 MB)
  float* ws   = (float*)d_ws;
  float* bufA = ws;
  float* bufB = bufA + (size_t)NTOK * K_;
  float* xseq = bufB + (size_t)NTOK * K_;
  float* zA   = xseq + (size_t)NTOK * K_;
  float* zBuf = zA   + (size_t)NTOK * M_;

  dim3 blk(128);
  dim3 gK(NTOK / 16, 4);   // Kout = 256
  dim3 gP(NTOK / 16, 2);   // Kout = 128
  dim3 gM(NTOK / 16, 1);   // Kout = 64

  // ---- x encoder: in_seq -> xseq ----
  gemm_bias_act<128><<<gK, blk, 0, stream>>>(in_seq, ex1w, ex1b, bufA, K_, 1);
  gemm_bias_act<256><<<gK, blk, 0, stream>>>(bufA,   ex2w, ex2b, bufB, K_, 1);
  gemm_bias_act<256><<<gK, blk, 0, stream>>>(bufB,   ex3w, ex3b, xseq, K_, 1);

  // ---- z encoder: in_seq -> z_seq (output region 3) ----
  gemm_bias_act<128><<<gM, blk, 0, stream>>>(in_seq, ez1w, ez1b, zA,    M_, 1);
  gemm_bias_act<64> <<<gM, blk, 0, stream>>>(zA,     ez2w, ez2b, zBuf,  M_, 1);
  gemm_bias_act<64> <<<gM, blk, 0, stream>>>(zBuf,   ez3w, ez3b, z_seq, M_, 1);

  // ---- x_recon = dec(xseq) (output region 1) ----
  gemm_bias_act<256><<<gK, blk, 0, stream>>>(xseq, de1w, de1b, bufA,    K_, 1);
  gemm_bias_act<256><<<gP, blk, 0, stream>>>(bufA, de2w, de2b, x_recon, P_, 1);

  // ---- serial scan: xhist -> bufB, z_pred -> output region 2 ----
  scan_kernel<<<dim3(B_), dim3(32), 0, stream>>>(xseq, z_seq, Wz, Azw, Azb, Lm, Rm,
                                                 bufB, z_pred);

  // ---- x_pred = dec(xhist) (output region 0) ----
  gemm_bias_act<256><<<gK, blk, 0, stream>>>(bufB, de1w, de1b, bufA,   K_, 1);
  gemm_bias_act<256><<<gP, blk, 0, stream>>>(bufA, de2w, de2b, x_pred, P_, 1);
}